// BaseNeuralRender_42468636622973
// MI455X (gfx1250) — compile-verified
//
#include <hip/hip_runtime.h>
#include <stdint.h>

#ifndef __has_builtin
#define __has_builtin(x) 0
#endif

typedef float v2f __attribute__((ext_vector_type(2)));
typedef float v8f __attribute__((ext_vector_type(8)));
typedef int b128v __attribute__((vector_size(4 * sizeof(int))));  // matches builtin proto

namespace {
constexpr int kSC      = 64;          // coarse samples (dists per ray)
constexpr int kSCW     = 63;          // weights per ray
constexpr int kSF      = 128;         // fine samples per ray
constexpr int kOut     = kSC + kSF;   // 192 outputs per ray
constexpr int kCStride = kSC + 1;     // cdf stride (padded: kills guard + bank conflicts)
constexpr int kRPB     = 32;          // rays per block
constexpr int kThreads = 256;         // 8 wave32
}

#if __has_builtin(__builtin_amdgcn_global_load_async_to_lds_b128) && \
    __has_builtin(__builtin_amdgcn_s_wait_asynccnt)
#define HAVE_ASYNC 1
#else
#define HAVE_ASYNC 0
#endif

#if __has_builtin(__builtin_amdgcn_global_store_async_from_lds_b128)
#define HAVE_ASYNC_ST 1
#else
#define HAVE_ASYNC_ST 0
#endif

#if __has_builtin(__builtin_amdgcn_wmma_f32_16x16x4_f32)
#define HAVE_WMMA_F32 1
#else
#define HAVE_WMMA_F32 0
#endif

__device__ __forceinline__ __attribute__((address_space(1))) b128v* as_g128(const float* p) {
  return (__attribute__((address_space(1))) b128v*)(uintptr_t)p;
}
__device__ __forceinline__ __attribute__((address_space(3))) b128v* as_l128(const float* p) {
  // Flat LDS pointer's low 32 bits are the LDS byte offset (ISA 10.2).
  return (__attribute__((address_space(3))) b128v*)(uint32_t)(uintptr_t)p;
}

// Stage nfloats (multiple of 4, 16B-aligned both sides) from global to LDS.
__device__ __forceinline__ void g2lds(const float* __restrict__ g, float* l,
                                      int nfloats, int tid) {
  const int nvec = nfloats >> 2;
#if HAVE_ASYNC
  for (int i = tid; i < nvec; i += kThreads) {
    __builtin_amdgcn_global_load_async_to_lds_b128(as_g128(g + 4 * i), as_l128(l + 4 * i), 0, 0);
  }
#else
  const float4* gv = reinterpret_cast<const float4*>(g);
  float4* lv = reinterpret_cast<float4*>(l);
  for (int i = tid; i < nvec; i += kThreads) lv[i] = gv[i];
#endif
}

__global__ __launch_bounds__(kThreads)
void nerf_fine_sample_kernel(const float* __restrict__ dists,
                             const float* __restrict__ weights,
                             const float* __restrict__ urand,
                             float* __restrict__ out) {
  __shared__ __align__(16) float s_d[kRPB * kSC];           // [ray][k]   coarse dists
  __shared__ __align__(16) float s_w[kRPB * kSCW + 4];      // raw weights, linear
  __shared__ __align__(16) float s_pdf[kSC * kRPB];         // [k][ray]   transposed pdf (row 63 = 0)
  __shared__ __align__(16) float s_cdf[kRPB * kCStride];    // [ray][k]   cdf, stride 65, slot 64 = pad
  __shared__ __align__(16) float s_u[kRPB * kSF];           // u -> sorted u -> sorted samples
  __shared__ __align__(16) float s_o[kRPB * kOut];          // merged output staging
  __shared__ float s_sum[kRPB];

  const int tid = threadIdx.x;
  const int blk = blockIdx.x;

  // ---- Phase 0: async-stage this block's rays into LDS ----
  g2lds(dists   + (size_t)blk * kRPB * kSC,  s_d, kRPB * kSC,  tid);
  g2lds(weights + (size_t)blk * kRPB * kSCW, s_w, kRPB * kSCW, tid);
  g2lds(urand   + (size_t)blk * kRPB * kSF,  s_u, kRPB * kSF,  tid);
#if HAVE_ASYNC
  __builtin_amdgcn_s_wait_asynccnt(0);
#endif
  __syncthreads();

  // ---- Phase 1: per-ray sum of (w + 0.01) ----
  if (tid < kRPB) {
    float s = 0.f;
    const float* wr = s_w + tid * kSCW;
    for (int k = 0; k < kSCW; ++k) s += wr[k] + 0.01f;
    s_sum[tid] = s;
  }
  __syncthreads();

  // ---- Phase 2: transposed pdf [k][ray], padded row 63 = 0 ----
  for (int idx = tid; idx < kSC * kRPB; idx += kThreads) {
    const int k = idx >> 5, ray = idx & 31;
    s_pdf[idx] = (k < kSCW) ? ((s_w[ray * kSCW + k] + 0.01f) / s_sum[ray]) : 0.0f;
  }
  __syncthreads();

  // ---- Phase 3: CDF = L(64x64 lower-tri ones) x P(64x32) via WMMA f32 16x16x4 ----
  // All 4 k-tiles run unconditionally: A-tiles above the diagonal are all-zero
  // (k0 > rowm => a = 0), contributing exactly +0.0 -- no divergent loop needed.
#if HAVE_WMMA_F32
  {
    const int lane  = tid & 31;
    const int wave  = tid >> 5;
    const int rayg  = wave & 1;    // 16-ray group (0 or 1)
    const int rtile = wave >> 1;   // output row tile 0..3
    const int m     = lane & 15;
    const int half  = lane >> 4;   // f32 A/B: lanes 0-15 -> K{0,1}, lanes 16-31 -> K{2,3}
    const int rowm  = rtile * 16 + m;
    v8f acc = {0.f, 0.f, 0.f, 0.f, 0.f, 0.f, 0.f, 0.f};
#pragma unroll
    for (int kt = 0; kt < 4; ++kt) {
#pragma unroll
      for (int kk = 0; kk < 4; ++kk) {
        const int k0 = kt * 16 + kk * 4 + half * 2;
        v2f a, b;
        a.x = (k0     <= rowm) ? 1.f : 0.f;           // L[rowm][k0]
        a.y = (k0 + 1 <= rowm) ? 1.f : 0.f;           // L[rowm][k0+1]
        b.x = s_pdf[k0 * kRPB + rayg * 16 + m];       // P[k0][ray]
        b.y = s_pdf[(k0 + 1) * kRPB + rayg * 16 + m]; // P[k0+1][ray]
        acc = __builtin_amdgcn_wmma_f32_16x16x4_f32(
            false, a, false, b, (short)0, acc, false, false);
      }
    }
    const int ray = rayg * 16 + m;
#pragma unroll
    for (int v = 0; v < 8; ++v) {
      const int row = rtile * 16 + v + half * 8;      // C: M = v + 8*half
      s_cdf[ray * kCStride + row + 1] = acc[v];       // row 63 lands in pad slot 64
    }
  }
#else
  if (tid < kRPB) {
    float c = 0.f;
    for (int k = 0; k < kSC - 1; ++k) {
      c += s_pdf[k * kRPB + tid];
      s_cdf[tid * kCStride + k + 1] = c;
    }
  }
#endif
  if (tid < kRPB) s_cdf[tid * kCStride] = 0.f;

  // ---- Phase 4: bitonic sort of each ray's 128 uniforms (monotone map => sorted samples) ----
  for (int size = 2; size <= kSF; size <<= 1) {
    for (int stride = size >> 1; stride > 0; stride >>= 1) {
      __syncthreads();
      for (int c = tid; c < kRPB * (kSF / 2); c += kThreads) {
        const int ray = c >> 6;         // 64 comparators per ray
        const int j   = c & 63;
        const int low = j & (stride - 1);
        const int i1  = ((j ^ low) << 1) | low;
        const int i2  = i1 + stride;
        float* base = s_u + ray * kSF;
        const float a  = base[i1], bb = base[i2];
        const float lo = fminf(a, bb), hi = fmaxf(a, bb);
        const bool asc = (i1 & size) == 0;
        base[i1] = asc ? lo : hi;
        base[i2] = asc ? hi : lo;
      }
    }
  }
  __syncthreads();

  // ---- Phase 5: inverse-CDF map sorted u -> sorted samples (in place) ----
  for (int f = tid; f < kRPB * kSF; f += kThreads) {
    const int ray = f >> 7;
    const float u = s_u[f];
    const float* c = s_cdf + ray * kCStride;
    int p = 0;  // largest index with c[p] <= u  (c[0]=0 <= u always)
#pragma unroll
    for (int st = 32; st >= 1; st >>= 1) {
      const int n = p + st;
      if (n < kSC && c[n] <= u) p = n;
    }
    const int below = p;
    const int above = (p + 1 < kSC) ? p + 1 : kSC - 1;
    const float c0 = c[below], c1 = c[above];
    float denom = c1 - c0;
    if (denom < 1e-5f) denom = 1.0f;
    const float t = (u - c0) / denom;
    const float* d = s_d + ray * kSC;
    const float d0 = d[below], d1 = d[above];
    s_u[f] = d0 + t * (d1 - d0);
  }
  __syncthreads();

  // ---- Phase 6: merge sorted samples (128) with sorted dists (64) by rank, into LDS staging ----
  for (int f = tid; f < kRPB * kOut; f += kThreads) {
    const int ray = f / kOut;
    const int e   = f - ray * kOut;
    const float* s = s_u + ray * kSF;
    const float* d = s_d + ray * kSC;
    float v;
    int pos;
    if (e < kSF) {                 // sample element: rank = e + #{dists < v}
      v = s[e];
      int cnt = 0;
#pragma unroll
      for (int st = 64; st >= 1; st >>= 1) {
        const int n = cnt + st;
        if (n <= kSC && d[n - 1] < v) cnt = n;
      }
      pos = e + cnt;
    } else {                       // dist element: rank = j + #{samples <= v}
      const int j = e - kSF;
      v = d[j];
      int cnt = 0;
#pragma unroll
      for (int st = 128; st >= 1; st >>= 1) {
        const int n = cnt + st;
        if (n <= kSF && s[n - 1] <= v) cnt = n;
      }
      pos = j + cnt;
    }
    s_o[ray * kOut + pos] = v;
  }
  __syncthreads();

  // ---- Phase 7: drain staged output with coalesced 16B async stores ----
  float* gout = out + (size_t)blk * kRPB * kOut;
#if HAVE_ASYNC_ST
  for (int i = tid; i < (kRPB * kOut) >> 2; i += kThreads) {
    __builtin_amdgcn_global_store_async_from_lds_b128(as_g128(gout + 4 * i),
                                                      as_l128(s_o + 4 * i), 0, 0);
  }
  // S_ENDPGM performs an implicit wait-idle; ASYNCcnt drains before wave exit.
#else
  float4* gv = reinterpret_cast<float4*>(gout);
  const float4* lv = reinterpret_cast<const float4*>(s_o);
  for (int i = tid; i < (kRPB * kOut) >> 2; i += kThreads) gv[i] = lv[i];
#endif
}

extern "C" void kernel_launch(void* const* d_in, const int* in_sizes, int n_in,
                              void* d_out, int out_size, void* d_ws, size_t ws_size,
                              hipStream_t stream) {
  const float* dists   = (const float*)d_in[0];  // [B][64]
  const float* weights = (const float*)d_in[1];  // [B][63]
  const float* urand   = (const float*)d_in[2];  // [B][128]
  float* out = (float*)d_out;                    // [B][192]
  const int n_rays = in_sizes[0] / kSC;          // 262144
  const int blocks = n_rays / kRPB;              // 8192 (B divisible by 32)
  nerf_fine_sample_kernel<<<blocks, kThreads, 0, stream>>>(dists, weights, urand, out);
}